// fofe_flex_all_encoder_79001628443160
// MI455X (gfx1250) — compile-verified
//
#include <hip/hip_runtime.h>

typedef float v2f __attribute__((ext_vector_type(2)));
typedef float v8f __attribute__((ext_vector_type(8)));

#define LFIX  512   // L
#define NWIN  49    // windows 0..47 plus 64
#define TLEN  513   // L + 1
#define LASTW 64

__global__ __launch_bounds__(256)
void fofe_flex_all_encoder_kernel(const float* __restrict__ x,
                                  const float* __restrict__ alpha,
                                  float* __restrict__ out,
                                  int C) {
    __shared__ float xs[LFIX];
    __shared__ float zpad[LASTW + 1 + LFIX]; // zpad[k] = z[k-65]; zpad[0..64] = 0
    __shared__ float apow[LASTW + 1];        // alpha^0 .. alpha^64
    __shared__ float carr[32];               // carry into each 16-tile

    const int row = blockIdx.x;              // b*C + c
    const int tid = threadIdx.x;
    const int c   = row % C;
    const float al = alpha[c];

    // stage x row; zero the pad; build alpha-power table
    xs[tid]       = x[(size_t)row * LFIX + tid];
    xs[tid + 256] = x[(size_t)row * LFIX + tid + 256];
    if (tid < LASTW + 1) zpad[tid] = 0.0f;
    if (tid == 0) {
        float p = 1.0f;
        for (int k = 0; k <= LASTW; ++k) { apow[k] = p; p *= al; }
    }
    __syncthreads();

    // ---- tile-local scans as triangular matmul: ZL = T * X via f32 WMMA ----
    // T[m,k] = (m>=k) ? alpha^(m-k) : 0   (16x16, chained as 4 K=4 steps)
    // X[k,n] = xs[16*n + k]               (n = tile index, 2 groups of 16)
    if (tid < 32) {
        const int lane = tid;
        const int m  = lane & 15;       // M row (A) / N col (B,D)
        const int hi = lane >> 4;       // selects K pair {0,1} vs {2,3}
        v2f a[4];
        #pragma unroll
        for (int kk = 0; kk < 4; ++kk) {
            const int k0 = kk * 4 + hi * 2;
            a[kk].x = (m >= k0)     ? apow[m - k0]     : 0.0f;
            a[kk].y = (m >= k0 + 1) ? apow[m - k0 - 1] : 0.0f;
        }
        #pragma unroll
        for (int g = 0; g < 2; ++g) {
            v8f acc = {};
            #pragma unroll
            for (int kk = 0; kk < 4; ++kk) {
                v2f b;
                const int base = 16 * (g * 16 + m) + kk * 4 + hi * 2;
                b.x = xs[base];
                b.y = xs[base + 1];
                acc = __builtin_amdgcn_wmma_f32_16x16x4_f32(
                    false, a[kk], false, b, (short)0, acc, false, false);
            }
            #pragma unroll
            for (int r = 0; r < 8; ++r) {
                const int mloc = r + 8 * hi;          // local time within tile
                const int tile = g * 16 + m;          // tile index (N)
                zpad[65 + tile * 16 + mloc] = acc[r]; // local scan, pre-carry
            }
        }
    }
    __syncthreads();

    // ---- serial carry chain across 32 tiles (31 fma, trivial) ----
    if (tid == 0) {
        float cv = 0.0f;
        carr[0] = 0.0f;
        const float a16 = apow[16];
        for (int j = 1; j < 32; ++j) {
            cv = zpad[65 + (j - 1) * 16 + 15] + a16 * cv; // z[16j-1]
            carr[j] = cv;
        }
    }
    __syncthreads();

    // ---- apply carries: z[16j+t'] = zl[t'] + alpha^(t'+1) * carr[j] ----
    for (int idx = tid; idx < LFIX; idx += 256) {
        const int j  = idx >> 4;
        const int tt = idx & 15;
        zpad[65 + idx] += apow[tt + 1] * carr[j];
    }
    __syncthreads();

    // ---- streaming write phase: out[w,t] = z[t-1] - alpha^w * z[t-1-w] ----
    const float a64 = apow[LASTW];
    float* outRow = out + (size_t)row * NWIN * TLEN;
    for (int t = tid; t < TLEN; t += 256) {
        const float base = zpad[LASTW + t];  // z[t-1], reused across 49 windows
        float* op = outRow + t;              // lane-consecutive t -> coalesced
        float nap = -1.0f;                   // -alpha^iw, updated by one v_mul
        #pragma unroll
        for (int iw = 0; iw < 48; ++iw) {
            const float sh = zpad[t + (LASTW - iw)];           // z[t-1-iw]
            __builtin_nontemporal_store(fmaf(nap, sh, base), op + iw * TLEN);
            nap *= al;
        }
        // window w = 64
        __builtin_nontemporal_store(fmaf(-a64, zpad[t], base), op + 48 * TLEN);
    }
}

extern "C" void kernel_launch(void* const* d_in, const int* in_sizes, int n_in,
                              void* d_out, int out_size, void* d_ws, size_t ws_size,
                              hipStream_t stream) {
    const float* x     = (const float*)d_in[0];
    const float* alpha = (const float*)d_in[1];
    float* out = (float*)d_out;
    const int C    = in_sizes[1];          // 512 (alpha has C elements)
    const int rows = in_sizes[0] / LFIX;   // B*C = 4096
    fofe_flex_all_encoder_kernel<<<rows, 256, 0, stream>>>(x, alpha, out, C);
}